// CacheUpdateAndAttend_85856396247835
// MI455X (gfx1250) — compile-verified
//
#include <hip/hip_runtime.h>

// ---------------- problem constants (from reference) ----------------
#define BB    32      // batch
#define HH    32      // query heads
#define HKV   8       // kv heads
#define GQ    4       // H / HKV
#define DD    128     // head dim
#define PP    32      // logical pages per sequence
#define PS    128     // slots per page
#define NP    (BB*PP) // physical pages

#define NT    256     // threads per workgroup (8 waves)
#define NW    8       // waves per workgroup

typedef __attribute__((ext_vector_type(2))) float v2f;
typedef __attribute__((ext_vector_type(4))) float v4f;
typedef __attribute__((ext_vector_type(8))) float v8f;

// -------------------------------------------------------------------
// Single fused pass: for each (b, kv-head) stream that sequence's 32
// pages exactly once from HBM: patch the new K/V row, write the updated
// cache copy (non-temporal), stage the page in LDS, and run flash-decode
// attention (f32 WMMA) on the causally-visible part. One HBM read + one
// HBM write per cache byte = bandwidth roofline.
// -------------------------------------------------------------------
__global__ __launch_bounds__(NT) void fused_paged_attn_kernel(
    const float* __restrict__ q,      // [B,H,1,D]
    const float* __restrict__ key,    // [B,HKV,1,D]
    const float* __restrict__ val,    // [B,HKV,1,D]
    const float* __restrict__ kin,    // [NP,HKV,PS,D] original K cache
    const float* __restrict__ vin,    // [NP,HKV,PS,D] original V cache
    const int*   __restrict__ pos_a,  // [B]
    const int*   __restrict__ ptab,   // [B,P]
    float* __restrict__ out,          // [B,H,1,D]
    float* __restrict__ kout,         // updated K cache (output)
    float* __restrict__ vout)         // updated V cache (output)
{
  const int blk  = blockIdx.x;       // 0..255
  const int b    = blk >> 3;
  const int kh   = blk & 7;
  const int tid  = threadIdx.x;
  const int w    = tid >> 5;         // wave id 0..7 (wave32)
  const int lane = tid & 31;
  const int n    = lane & 15;
  const int h    = lane >> 4;
  const int nq   = (n < 4) ? n : 3;  // clamped lane for Q / P operands

  __shared__ float skx[PS * DD];      // 64KB staged K page slice
  __shared__ float svx[PS * DD];      // 64KB staged V page slice
  __shared__ float qs [GQ * DD];      // Q rows for this kv head
  __shared__ float plds[NW * 64];     // per-wave P tile [g][l]
  __shared__ float macc[NW * GQ * DD];
  __shared__ float mbuf[NW * GQ];
  __shared__ float sbuf[NW * GQ];

  const int pos     = pos_a[b];
  const int pospage = pos >> 7;
  const int slot    = pos & (PS - 1);

  for (int i = tid; i < GQ * DD; i += NT)
    qs[i] = q[(size_t)(b * HH + kh * GQ) * DD + i];
  // (covered by the first in-loop barrier: page 0 always needs attention)

  const float scale  = 0.088388347648318447f; // 1/sqrt(128)
  const float NEG    = -__builtin_inff();

  v8f acc[8] = {};                    // 8 d-tiles of C-matrices (rows 0..3 valid)
  float m = NEG, s = 0.f;

  float*       pw     = &plds[w * 64];
  const float* qrow   = &qs[nq * DD + 2 * h];
  const int    sblock = w * 16;       // this wave's 16-key tile within every page

  const float* keyrow = key + (size_t)(b * HKV + kh) * DD;
  const float* valrow = val + (size_t)(b * HKV + kh) * DD;

  for (int p = 0; p < PP; ++p) {
    const int    phys  = ptab[b * PP + p];
    const size_t pbase = (size_t)(phys * HKV + kh) * (PS * DD);
    const v4f* ksrc = (const v4f*)(kin + pbase);
    const v4f* vsrc = (const v4f*)(vin + pbase);
    v4f* kdst = (v4f*)(kout + pbase);
    v4f* vdst = (v4f*)(vout + pbase);
    v4f* kl = (v4f*)skx;
    v4f* vl = (v4f*)svx;

    const bool patch = (p == pospage);        // this page holds the new row
    const bool need  = (p * PS) <= pos;       // any causally-visible keys?

    // ---- stream page: HBM read-once -> (patch) -> HBM write + LDS stage ----
    #pragma unroll 4
    for (int i = tid; i < PS * DD / 4; i += NT) {
      const int  row = i >> 5;                // 32 v4f per 128-float row
      const int  d4  = i & 31;
      const bool pk  = patch && (row == slot);
      v4f kv = pk ? *(const v4f*)(keyrow + 4 * d4)
                  : __builtin_nontemporal_load(&ksrc[i]);
      v4f vv = pk ? *(const v4f*)(valrow + 4 * d4)
                  : __builtin_nontemporal_load(&vsrc[i]);
      __builtin_nontemporal_store(kv, &kdst[i]);
      __builtin_nontemporal_store(vv, &vdst[i]);
      if (need) { kl[i] = kv; vl[i] = vv; }
    }

    // pull next page toward L2 so HBM stays busy during the compute phase
    if (p + 1 < PP) {
      const int    phys2 = ptab[b * PP + p + 1];
      const size_t nb    = (size_t)(phys2 * HKV + kh) * (PS * DD);
      #pragma unroll
      for (int j = 0; j < 4; ++j) {
        __builtin_prefetch(kin + nb + (size_t)(tid + j * NT) * 32, 0, 0);
        __builtin_prefetch(vin + nb + (size_t)(tid + j * NT) * 32, 0, 0);
      }
    }

    if (!need) continue;                      // pure-copy page (uniform)
    __syncthreads();

    const int lbase0 = p * PS + sblock;       // first key index of this tile
    if (lbase0 <= pos) {                      // wave-uniform: tile has valid keys
      // ---- QK: D[l,g], 32 chained f32 WMMAs over d-chunks of 4 ----
      const float* krow = &skx[(sblock + n) * DD + 2 * h];
      v8f sc = {};
      #pragma unroll
      for (int c = 0; c < 32; ++c) {
        v2f a  = *(const v2f*)(krow + 4 * c);
        v2f bq = *(const v2f*)(qrow + 4 * c);
        sc = __builtin_amdgcn_wmma_f32_16x16x4_f32(false, a, false, bq,
                                                   (short)0, sc, false, false);
      }

      // scale + causal mask (element (r,h) is key l = lbase0 + r + 8h)
      const int lb = lbase0 + 8 * h;
      #pragma unroll
      for (int r = 0; r < 8; ++r) {
        const float vsc = sc[r] * scale;
        sc[r] = ((lb + r) <= pos) ? vsc : NEG;
      }

      // online softmax per column g (column g lives in lanes g, g+16)
      float tmax = sc[0];
      #pragma unroll
      for (int r = 1; r < 8; ++r) tmax = fmaxf(tmax, sc[r]);
      tmax = fmaxf(tmax, __shfl_xor(tmax, 16, 32));
      const float mnew  = fmaxf(m, tmax);
      const float alpha = __expf(m - mnew);
      m = mnew;

      v8f p8;
      float tsum = 0.f;
      #pragma unroll
      for (int r = 0; r < 8; ++r) { const float e = __expf(sc[r] - mnew); p8[r] = e; tsum += e; }
      tsum += __shfl_xor(tsum, 16, 32);
      s = s * alpha + tsum;

      const float a0 = __shfl(alpha, 0, 32);
      const float a1 = __shfl(alpha, 1, 32);
      const float a2 = __shfl(alpha, 2, 32);
      const float a3 = __shfl(alpha, 3, 32);
      #pragma unroll
      for (int dt = 0; dt < 8; ++dt) {
        acc[dt][0] *= a0; acc[dt][1] *= a1; acc[dt][2] *= a2; acc[dt][3] *= a3;
      }

      // bounce P through LDS: C-layout -> A-operand layout (intra-wave)
      if (n < 4) {
        #pragma unroll
        for (int r = 0; r < 8; ++r) pw[n * 16 + r + 8 * h] = p8[r];
      }
      asm volatile("s_wait_dscnt 0" ::: "memory");

      // ---- PV: out[g,d] += sum_l P[g,l] * V[l,d] ----
      const float* vrow = &svx[(sblock + 2 * h) * DD + n];
      const float* prow = &pw[nq * 16 + 2 * h];
      #pragma unroll
      for (int k0 = 0; k0 < 16; k0 += 4) {
        v2f pa = *(const v2f*)(prow + k0);
        #pragma unroll
        for (int dt = 0; dt < 8; ++dt) {
          v2f bv;
          bv.x = vrow[k0 * DD + dt * 16];
          bv.y = vrow[(k0 + 1) * DD + dt * 16];
          acc[dt] = __builtin_amdgcn_wmma_f32_16x16x4_f32(false, pa, false, bv,
                                                          (short)0, acc[dt], false, false);
        }
      }
    }
    __syncthreads();   // protect LDS before next page's staging
  }

  // ---- publish per-wave flash state and merge the 8 waves ----
  if (n < 4 && h == 0) { mbuf[w * GQ + n] = m; sbuf[w * GQ + n] = s; }
  if (h == 0) {
    #pragma unroll
    for (int dt = 0; dt < 8; ++dt)
      #pragma unroll
      for (int g = 0; g < 4; ++g)
        macc[(w * GQ + g) * DD + dt * 16 + n] = acc[dt][g];
  }
  __syncthreads();

  for (int idx = tid; idx < GQ * DD; idx += NT) {
    const int g = idx >> 7;
    const int d = idx & (DD - 1);
    float mm = NEG;
    for (int wv = 0; wv < NW; ++wv) mm = fmaxf(mm, mbuf[wv * GQ + g]);
    float ss = 0.f, oo = 0.f;
    for (int wv = 0; wv < NW; ++wv) {
      const float f = __expf(mbuf[wv * GQ + g] - mm);
      ss += sbuf[wv * GQ + g] * f;
      oo += macc[(wv * GQ + g) * DD + d] * f;
    }
    out[(size_t)(b * HH + kh * GQ + g) * DD + d] = oo / ss;
  }
}

// -------------------------------------------------------------------
extern "C" void kernel_launch(void* const* d_in, const int* in_sizes, int n_in,
                              void* d_out, int out_size, void* d_ws, size_t ws_size,
                              hipStream_t stream) {
  const float* q    = (const float*)d_in[0];
  const float* key  = (const float*)d_in[1];
  const float* val  = (const float*)d_in[2];
  const float* kci  = (const float*)d_in[3];
  const float* vci  = (const float*)d_in[4];
  const int*   pos  = (const int*)d_in[5];
  const int*   ptab = (const int*)d_in[6];

  float* out = (float*)d_out;
  const size_t out_elems   = (size_t)BB * HH * DD;        // 131072
  const size_t cache_elems = (size_t)NP * HKV * PS * DD;  // 134217728
  float* kout = out + out_elems;
  float* vout = kout + cache_elems;

  // one fused pass: copy+update caches and attend, each HBM byte touched once
  fused_paged_attn_kernel<<<BB * HKV, NT, 0, stream>>>(
      q, key, val, kci, vci, pos, ptab, out, kout, vout);
}